// UnimodalBeta_12678743458351
// MI455X (gfx1250) — compile-verified
//
#include <hip/hip_runtime.h>
#include <math.h>

// ---------------------------------------------------------------------------
// Stage 1: dual GEMV  sa = x@Wa, sb = x@Wb  via V_WMMA_F32_16X16X4_F32
//   - exact f32 accumulate (matches reference precision)
//   - A = 16-row tile of x, B = [Wa | Wb | 0 ... 0] K-chunk
//   - x staged global->LDS coalesced, LDS row stride padded to 68 floats
// Stage 2: per-row Beta quadrature (softplus/clip/lgamma + 550 exp per row)
//   with log(grid), log1p(-grid) tables precomputed into LDS.
// ---------------------------------------------------------------------------

typedef __attribute__((ext_vector_type(2))) float v2f;
typedef __attribute__((ext_vector_type(4))) float v4f;
typedef __attribute__((ext_vector_type(8))) float v8f;

#define CDIM 512
#define ROWS_PER_WAVE 16
#define WAVES_PER_WG 8
#define ROWS_PER_WG (ROWS_PER_WAVE * WAVES_PER_WG)   // 128
#define KCHUNK 64
#define XSTRIDE 68   // padded LDS row stride (floats): 68%64=4 -> conflict-free,
                     // 68*4B=272B is a multiple of 16B -> b128 stores OK

__global__ __launch_bounds__(256) void beta_gemv_wmma(
    const float* __restrict__ x, const float* __restrict__ Wa,
    const float* __restrict__ Wb, float* __restrict__ sab, int nrows)
{
    __shared__ float wlds[2 * CDIM];                              // interleaved (Wa,Wb)
    __shared__ float xlds[WAVES_PER_WG][ROWS_PER_WAVE * XSTRIDE]; // per-wave x tile

    const int tid   = threadIdx.x;
    const int wave  = tid >> 5;
    const int lane  = tid & 31;
    const int half  = lane >> 4;    // which 16-lane half
    const int lan16 = lane & 15;
    const int r0    = blockIdx.x * ROWS_PER_WG + wave * ROWS_PER_WAVE;

    for (int k = tid; k < CDIM; k += blockDim.x) {
        wlds[2 * k]     = Wa[k];
        wlds[2 * k + 1] = Wb[k];
    }
    __syncthreads();

    float* xs = &xlds[wave][0];
    v8f acc = {};

    for (int k0 = 0; k0 < CDIM; k0 += KCHUNK) {
        // Stage 16 rows x 64 cols of x into LDS. Each iteration: 16 lanes cover
        // one row's 64 floats as 16 float4s -> fully coalesced 256B per half-wave.
        #pragma unroll
        for (int it = 0; it < 8; ++it) {
            int row = it * 2 + half;
            v4f v = *(const v4f*)&x[(size_t)(r0 + row) * CDIM + k0 + lan16 * 4];
            *(v4f*)&xs[row * XSTRIDE + lan16 * 4] = v;
        }
        // Each wave touches only its own LDS tile; DS ops are in-order per wave,
        // so no workgroup barrier needed inside the K loop.

        #pragma unroll
        for (int kk = 0; kk < KCHUNK; kk += 4) {
            // A fragment (16x4 f32): v0 holds K=base+{0,2}, v1 K=base+{1,3},
            // split across lane halves; lane L<16 -> M=L.
            int klocal = kk + half * 2;
            v2f a = *(const v2f*)&xs[lan16 * XSTRIDE + klocal];

            // B fragment (4x16 f32), same K split convention as A.
            // Column N=0 is Wa, N=1 is Wb, others zero.
            int kglob = k0 + klocal;
            v2f w0 = *(const v2f*)&wlds[2 * kglob];        // (Wa[k],   Wb[k])
            v2f w1 = *(const v2f*)&wlds[2 * (kglob + 1)];  // (Wa[k+1], Wb[k+1])
            v2f b;
            b.x = (lan16 == 0) ? w0.x : ((lan16 == 1) ? w0.y : 0.0f);
            b.y = (lan16 == 0) ? w1.x : ((lan16 == 1) ? w1.y : 0.0f);

            // D = A*B + C, f32, exact accumulate
            acc = __builtin_amdgcn_wmma_f32_16x16x4_f32(
                /*neg_a=*/false, a, /*neg_b=*/false, b,
                /*c_mod=*/(short)0, acc, /*reuse_a=*/false, /*reuse_b=*/false);
        }
    }

    // C/D layout: VGPR j: lanes0-15 -> C[M=j][N=lane]; lanes16-31 -> C[M=j+8][N=lane-16]
    // sa for row M is at N=0 (lanes 0/16), sb at N=1 (lanes 1/17).
    #pragma unroll
    for (int j = 0; j < 8; ++j) {
        float v = acc[j];
        if      (lane == 0)  sab[r0 + j]             = v;  // sa, rows j
        else if (lane == 16) sab[r0 + 8 + j]         = v;  // sa, rows j+8
        else if (lane == 1)  sab[nrows + r0 + j]     = v;  // sb, rows j
        else if (lane == 17) sab[nrows + r0 + 8 + j] = v;  // sb, rows j+8
    }
}

// ---------------------------------------------------------------------------

#define KCLS 10
#define NACT 550          // sum_{i=1..10} 10*i active grid points
#define EPSF 1e-8f

__device__ __forceinline__ float softplusf(float v) {
    if (v > 20.0f)  return v;            // log1p(exp(v)) ~ v
    if (v < -20.0f) return __expf(v);    // log1p(exp(v)) ~ exp(v)
    return log1pf(__expf(v));
}

__global__ __launch_bounds__(256) void beta_posterior(
    const float* __restrict__ sab, const float* __restrict__ ba,
    const float* __restrict__ bb, float* __restrict__ out, int nrows)
{
    __shared__ float lg[NACT];   // log(grid[p])
    __shared__ float l1g[NACT];  // log1p(-grid[p])

    // Build the active-point tables. Flattened offsets: off(i) = 5*i*(i-1).
    for (int p = threadIdx.x; p < NACT; p += blockDim.x) {
        int i = 1;
        while (5 * i * (i + 1) <= p) ++i;      // find grid row: off(i) <= p < off(i+1)
        int j = p - 5 * i * (i - 1);
        int n = 10 * i;                        // NPTS * i / K
        float hi   = (float)i * 0.1f;          // i / K
        float step = (hi - 2.0f * EPSF) / (float)(n - 1);
        float g    = EPSF + step * (float)j;   // linspace(EPS, hi-EPS, n)[j]
        lg[p]  = __logf(g);
        l1g[p] = log1pf(-g);
    }
    __syncthreads();

    int r = blockIdx.x * blockDim.x + threadIdx.x;
    if (r >= nrows) return;

    float alpha = 1.0f + softplusf(sab[r] + ba[0]);
    float beta  = 1.0f + softplusf(sab[nrows + r] + bb[0]);
    alpha = fminf(fmaxf(alpha, 1.0f), 100.0f);
    beta  = fminf(fmaxf(beta,  1.0f), 100.0f);

    float lognorm = lgammaf(alpha) + lgammaf(beta) - lgammaf(alpha + beta);
    float am1 = alpha - 1.0f, bm1 = beta - 1.0f;

    float cdf[KCLS + 1];
    cdf[0] = 0.0f;                   // grid row 0 has zero active points
    int p = 0;
    for (int i = 1; i <= KCLS; ++i) {
        int n = 10 * i;
        float s = 0.0f;
        for (int j = 0; j < n; ++j, ++p) {
            float lp = fmaf(am1, lg[p], fmaf(bm1, l1g[p], -lognorm));
            s += __expf(lp);
        }
        cdf[i] = s * 1.0e-3f;        // dx = 1/(NPTS*K)
    }

    float pr[KCLS];
    float tot = 0.0f;
    #pragma unroll
    for (int i = 0; i < KCLS; ++i) { pr[i] = cdf[i + 1] - cdf[i]; tot += pr[i]; }
    float inv = 1.0f / tot;
    #pragma unroll
    for (int i = 0; i < KCLS; ++i) out[(size_t)r * KCLS + i] = pr[i] * inv;
}

// ---------------------------------------------------------------------------

extern "C" void kernel_launch(void* const* d_in, const int* in_sizes, int n_in,
                              void* d_out, int out_size, void* d_ws, size_t ws_size,
                              hipStream_t stream) {
    const float* x  = (const float*)d_in[0];
    const float* Wa = (const float*)d_in[1];
    const float* ba = (const float*)d_in[2];
    const float* Wb = (const float*)d_in[3];
    const float* bb = (const float*)d_in[4];
    float* out = (float*)d_out;
    float* sab = (float*)d_ws;            // [2][nrows] floats: sa then sb (512 KB)

    int nrows = in_sizes[0] / CDIM;       // 65536

    beta_gemv_wmma<<<nrows / ROWS_PER_WG, 256, 0, stream>>>(x, Wa, Wb, sab, nrows);
    beta_posterior<<<(nrows + 255) / 256, 256, 0, stream>>>(sab, ba, bb, out, nrows);
}